// BaselineGCN_64811056497274
// MI455X (gfx1250) — compile-verified
//
#include <hip/hip_runtime.h>

typedef __attribute__((ext_vector_type(2))) float v2f;
typedef __attribute__((ext_vector_type(8))) float v8f;

#define NNODES 51200
#define NEDGES 819200
#define FIN 116
#define HID 64
#define NGRAPH 512
#define NPG 100
#define BSTRIDE 80   // float2 per k-group: 64 used + 16 pad -> +32 banks per group

// ---------------- degree / normalization ----------------
__global__ void deg_init_kernel(float* __restrict__ deg) {
    int n = blockIdx.x * blockDim.x + threadIdx.x;
    if (n < NNODES) deg[n] = 1.0f;   // self loop
}

__global__ void deg_atomic_kernel(const long long* __restrict__ dst,
                                  float* __restrict__ deg) {
    int e = blockIdx.x * blockDim.x + threadIdx.x;
    if (e < NEDGES) atomicAdd(&deg[(int)dst[e]], 1.0f);
}

__global__ void dinv_kernel(float* __restrict__ deg) {
    int n = blockIdx.x * blockDim.x + threadIdx.x;
    if (n < NNODES) deg[n] = rsqrtf(deg[n]);
}

// ---------------- fp32 WMMA GEMM: T[N,64] = relu?(Hin[N,K]) @ W[K,64] ----------------
// Block: 256 threads = 8 waves; wave computes a 16x64 tile; grid.x = NNODES/128.
// W staged in LDS pre-paired in B-fragment layout: group g holds
// {W[2g][c], W[2g+1][c]} as one float2 -> each B fragment is a single ds_load_b64.
template <int K, bool RELU>
__global__ __launch_bounds__(256) void gcn_gemm_wmma(
    const float* __restrict__ Hin, const float* __restrict__ W,
    float* __restrict__ T) {
    __shared__ v2f sW2[(K / 2) * BSTRIDE];

    int tid = threadIdx.x;
    for (int idx = tid; idx < (K / 2) * HID; idx += 256) {
        int g = idx >> 6, c = idx & 63;
        v2f p = { W[(2 * g) * HID + c], W[(2 * g + 1) * HID + c] };
        sW2[g * BSTRIDE + c] = p;
    }
    __syncthreads();

    int wave = tid >> 5;
    int lane = tid & 31;
    int half = lane >> 4;        // 0: lanes 0-15, 1: lanes 16-31
    int m    = lane & 15;
    int rowBase = blockIdx.x * 128 + wave * 16;
    const float* hrow = Hin + (size_t)(rowBase + m) * (size_t)K;

    v8f c0 = {}, c1 = {}, c2 = {}, c3 = {};

#pragma unroll 4
    for (int k = 0; k < K; k += 4) {
        int kk = k + half * 2;                 // A/B VGPR pair layout (ISA 7.12.2)
        float a0 = hrow[kk];
        float a1 = hrow[kk + 1];
        if (RELU) { a0 = fmaxf(a0, 0.0f); a1 = fmaxf(a1, 0.0f); }
        v2f a = { a0, a1 };

        const v2f* wg = sW2 + (size_t)(kk >> 1) * BSTRIDE;   // group for rows kk, kk+1
        v2f b0 = wg[m];
        v2f b1 = wg[16 + m];
        v2f b2 = wg[32 + m];
        v2f b3 = wg[48 + m];

        c0 = __builtin_amdgcn_wmma_f32_16x16x4_f32(false, a, false, b0, (short)0, c0, false, false);
        c1 = __builtin_amdgcn_wmma_f32_16x16x4_f32(false, a, false, b1, (short)0, c1, false, false);
        c2 = __builtin_amdgcn_wmma_f32_16x16x4_f32(false, a, false, b2, (short)0, c2, false, false);
        c3 = __builtin_amdgcn_wmma_f32_16x16x4_f32(false, a, false, b3, (short)0, c3, false, false);
    }

    // C/D layout: VGPR r holds M = r (lanes 0-15) and M = 8+r (lanes 16-31), N = lane%16
#pragma unroll
    for (int r = 0; r < 8; ++r) {
        int orow = rowBase + half * 8 + r;
        float* trow = T + (size_t)orow * HID;
        trow[m]      = c0[r];
        trow[16 + m] = c1[r];
        trow[32 + m] = c2[r];
        trow[48 + m] = c3[r];
    }
}

// ---------------- agg = T * dinv^2 (self loop) + bias ----------------
__global__ void agg_init_kernel(const float* __restrict__ T,
                                const float* __restrict__ dinv,
                                const float* __restrict__ bias,
                                float* __restrict__ agg) {
    int t = blockIdx.x * blockDim.x + threadIdx.x;   // covers NNODES*HID exactly
    int n = t >> 6, j = t & 63;
    float di = dinv[n];
    agg[t] = T[t] * di * di + bias[j];
}

// ---------------- wave-per-edge gather/scale/scatter ----------------
__global__ void edge_scatter_kernel(const float* __restrict__ T,
                                    const long long* __restrict__ src,
                                    const long long* __restrict__ dst,
                                    const float* __restrict__ dinv,
                                    float* __restrict__ agg) {
    unsigned t = blockIdx.x * blockDim.x + threadIdx.x;
    unsigned e = t >> 5;
    int lane = t & 31;
    if (e >= NEDGES) return;
    int s = (int)src[e];
    int d = (int)dst[e];
    float nrm = dinv[s] * dinv[d];
    float2 v = ((const float2*)(T + (size_t)s * HID))[lane];
    float* ad = agg + (size_t)d * HID + lane * 2;
    atomicAdd(ad,     v.x * nrm);
    atomicAdd(ad + 1, v.y * nrm);
}

// ---------------- fused ReLU + mean pool (100 nodes per graph) ----------------
__global__ void pool_kernel(const float* __restrict__ agg, float* __restrict__ pooled) {
    int g = blockIdx.x;          // 512 graphs
    int j = threadIdx.x;         // 64 channels
    const float* base = agg + (size_t)g * NPG * HID + j;
    float sum = 0.0f;
    for (int i = 0; i < NPG; ++i) sum += fmaxf(base[i * HID], 0.0f);
    pooled[g * HID + j] = sum * (1.0f / NPG);
}

// ---------------- head: feat = relu(pooled @ W_lin1 + b_lin1) ----------------
__global__ void head1_kernel(const float* __restrict__ pooled,
                             const float* __restrict__ Wl1,
                             const float* __restrict__ bl1,
                             float* __restrict__ feat) {
    int g = blockIdx.x;
    int f = threadIdx.x;         // 128 threads, 116 active
    if (f >= FIN) return;
    const float* p = pooled + g * HID;
    float acc = bl1[f];
    for (int j = 0; j < HID; ++j) acc += p[j] * Wl1[j * FIN + f];
    feat[g * FIN + f] = fmaxf(acc, 0.0f);
}

// ---------------- head: out = feat @ W_lin2 + b_lin2 ----------------
__global__ void head2_kernel(const float* __restrict__ feat,
                             const float* __restrict__ Wl2,
                             const float* __restrict__ bl2,
                             float* __restrict__ out) {
    int t = blockIdx.x * blockDim.x + threadIdx.x;
    if (t >= NGRAPH * 2) return;
    int g = t >> 1, c = t & 1;
    const float* fr = feat + g * FIN;
    float acc = bl2[c];
    for (int f = 0; f < FIN; ++f) acc += fr[f] * Wl2[f * 2 + c];
    out[t] = acc;
}

extern "C" void kernel_launch(void* const* d_in, const int* in_sizes, int n_in,
                              void* d_out, int out_size, void* d_ws, size_t ws_size,
                              hipStream_t stream) {
    const float*     x     = (const float*)d_in[0];
    const long long* eidx  = (const long long*)d_in[1];   // int64 [2, E]
    const long long* esrc  = eidx;
    const long long* edst  = eidx + NEDGES;
    // d_in[2] = batch (derivable: node/100), unused
    const float* W0  = (const float*)d_in[3];
    const float* b0  = (const float*)d_in[4];
    const float* W1  = (const float*)d_in[5];
    const float* b1  = (const float*)d_in[6];
    const float* W2  = (const float*)d_in[7];
    const float* b2  = (const float*)d_in[8];
    const float* Wl1 = (const float*)d_in[9];
    const float* bl1 = (const float*)d_in[10];
    const float* Wl2 = (const float*)d_in[11];
    const float* bl2 = (const float*)d_in[12];
    float* out = (float*)d_out;

    // workspace layout (floats)
    float* ws    = (float*)d_ws;
    float* dinv  = ws;                                   // [N]
    float* T     = dinv + NNODES;                        // [N,64]
    float* hA    = T    + (size_t)NNODES * HID;          // [N,64]
    float* hB    = hA   + (size_t)NNODES * HID;          // [N,64]
    float* pooled= hB   + (size_t)NNODES * HID;          // [512,64]
    float* feat  = pooled + NGRAPH * HID;                // [512,116]

    const int TB = 256;

    // normalization: deg = 1 + incoming edges; dinv = rsqrt(deg)
    deg_init_kernel<<<NNODES / TB, TB, 0, stream>>>(dinv);
    deg_atomic_kernel<<<NEDGES / TB, TB, 0, stream>>>(edst, dinv);
    dinv_kernel<<<NNODES / TB, TB, 0, stream>>>(dinv);

    const int gemmGrid    = NNODES / 128;                   // 400
    const int aggGrid     = (NNODES * HID) / TB;            // 12800
    const int scatterGrid = (NEDGES * 32) / TB;             // 102400

    // Layer 0: x[N,116] -> hA
    gcn_gemm_wmma<FIN, false><<<gemmGrid, TB, 0, stream>>>(x, W0, T);
    agg_init_kernel<<<aggGrid, TB, 0, stream>>>(T, dinv, b0, hA);
    edge_scatter_kernel<<<scatterGrid, TB, 0, stream>>>(T, esrc, edst, dinv, hA);

    // Layer 1: relu(hA)[N,64] -> hB
    gcn_gemm_wmma<HID, true><<<gemmGrid, TB, 0, stream>>>(hA, W1, T);
    agg_init_kernel<<<aggGrid, TB, 0, stream>>>(T, dinv, b1, hB);
    edge_scatter_kernel<<<scatterGrid, TB, 0, stream>>>(T, esrc, edst, dinv, hB);

    // Layer 2: relu(hB)[N,64] -> hA (reused)
    gcn_gemm_wmma<HID, true><<<gemmGrid, TB, 0, stream>>>(hB, W2, T);
    agg_init_kernel<<<aggGrid, TB, 0, stream>>>(T, dinv, b2, hA);
    edge_scatter_kernel<<<scatterGrid, TB, 0, stream>>>(T, esrc, edst, dinv, hA);

    // relu + mean pool, then MLP head
    pool_kernel<<<NGRAPH, HID, 0, stream>>>(hA, pooled);
    head1_kernel<<<NGRAPH, 128, 0, stream>>>(pooled, Wl1, bl1, feat);
    head2_kernel<<<(NGRAPH * 2 + TB - 1) / TB, TB, 0, stream>>>(feat, Wl2, bl2, out);
}